// GenODE_52965536694609
// MI455X (gfx1250) — compile-verified
//
#include <hip/hip_runtime.h>
#include <hip/hip_bf16.h>
#include <math.h>

typedef __attribute__((ext_vector_type(2))) float v2f;
typedef __attribute__((ext_vector_type(8))) float v8f;

#define DDIM 64
#define HDIM 128
#define NDRAWS 8192
#define TPTS 64
#define WAVES 4
#define BLOCK (WAVES * 32)            // 128 threads = 4 wave32
#define ROWS_PER_BLOCK (WAVES * 16)   // 64 rows of z per block
#define NBLOCKS (NDRAWS / ROWS_PER_BLOCK)

#define NT1 8   // N-tiles of 16 for GEMM1 (N = H = 128)
#define NT2 4   // N-tiles of 16 for GEMM2 (N = D = 64)
#define KC1 16  // K-chunks of 4 for GEMM1 (K = D = 64)
#define KC2 32  // K-chunks of 4 for GEMM2 (K = H = 128)

// V_WMMA_F32_16X16X4_F32 fragment layouts (ISA 7.12.2):
//   A 16x4 : lane l, elem e -> M = l&15, K = 2*(l>>4) + e   (contiguous f32 pair)
//   B 4x16 : lane l, elem e -> N = l&15, K = 2*(l>>4) + e   (contiguous f32 pair)
//   C 16x16: lane l, reg  r -> M = r + 8*(l>>4), N = l&15
// => fragments are plain 8-byte loads from row-major [rows][K] storage.

__global__ __launch_bounds__(BLOCK)
void GenODE_node_rk4_wmma_f32(const float* __restrict__ rand_e,
                              const float* __restrict__ z0_mean,
                              const float* __restrict__ z0_log_sigma,
                              const float* __restrict__ W1,
                              const float* __restrict__ b1,
                              const float* __restrict__ W2,
                              const float* __restrict__ b2,
                              float* __restrict__ out) {
    // Row-major fp32 LDS copies; WMMA f32 fragments load directly from these.
    __shared__ float w1s[HDIM * DDIM];          // 32 KB  [H][D]
    __shared__ float w2s[DDIM * HDIM];          // 32 KB  [D][H]
    __shared__ float b1s[HDIM];
    __shared__ float b2s[DDIM];
    __shared__ float zsts[WAVES * 16 * DDIM];   // 16 KB  per-wave [16][D] stage
    __shared__ float hsts[WAVES * 16 * HDIM];   // 32 KB  per-wave [16][H] stage

    const int tid  = threadIdx.x;
    const int lane = tid & 31;
    const int wave = tid >> 5;
    const int ncol = lane & 15;          // C-fragment N index
    const int mrow = (lane >> 4) << 3;   // C-fragment M base (0 or 8)
    const int koff = (lane >> 4) << 1;   // A/B-fragment K base (0 or 2)

    for (int idx = tid; idx < HDIM * DDIM; idx += BLOCK) w1s[idx] = W1[idx];
    for (int idx = tid; idx < DDIM * HDIM; idx += BLOCK) w2s[idx] = W2[idx];
    if (tid < HDIM) b1s[tid] = b1[tid];
    if (tid < DDIM) b2s[tid] = b2[tid];
    __syncthreads();

    const float sigma   = expf(z0_log_sigma[0]);
    const int   rowbase = blockIdx.x * ROWS_PER_BLOCK + wave * 16;
    float* zw = &zsts[wave * 16 * DDIM];
    float* hw = &hsts[wave * 16 * HDIM];

    // z state, RK accumulator, stage derivative: C-fragments (4 col-tiles x v8f)
    v8f z[NT2], acc[NT2], kd[NT2];

    // ---- z0 = z0_mean + sigma * rand_e ; write z0 region and z[t=0] ----
    for (int t = 0; t < NT2; ++t) {
        for (int r = 0; r < 8; ++r) {
            int row = rowbase + mrow + r;
            int col = 16 * t + ncol;
            float v = z0_mean[col] + sigma * rand_e[row * DDIM + col];
            z[t][r] = v;
            out[row * DDIM + col] = v;                               // z0
            out[NDRAWS * DDIM + TPTS + row * DDIM + col] = v;        // z[0]
        }
    }
    if (blockIdx.x == 0 && tid < TPTS)
        out[NDRAWS * DDIM + tid] = (float)tid * (1.0f / (TPTS - 1)); // t_span

    // stage = z + scale*k -> wave-private row-major [16][D] staging
    auto stage_store = [&](const v8f* zz, const v8f* kk, float scale) {
        for (int t = 0; t < NT2; ++t)
            for (int r = 0; r < 8; ++r)
                zw[(mrow + r) * DDIM + 16 * t + ncol] = zz[t][r] + scale * kk[t][r];
    };

    // one vector-field evaluation: vf(stage) = W2 tanh(W1 stage + b1) + b2
    auto vf = [&](v8f* res) {
        __syncthreads();
        // GEMM1: h_pre = stage @ W1^T   [16 x 128], fp32 WMMA, K-chunks of 4
        v8f hc[NT1] = {v8f{}, v8f{}, v8f{}, v8f{}, v8f{}, v8f{}, v8f{}, v8f{}};
        for (int kc = 0; kc < KC1; ++kc) {
            v2f a = *(const v2f*)&zw[ncol * DDIM + 4 * kc + koff];
            for (int t = 0; t < NT1; ++t) {
                v2f b = *(const v2f*)&w1s[(16 * t + ncol) * DDIM + 4 * kc + koff];
                hc[t] = __builtin_amdgcn_wmma_f32_16x16x4_f32(false, a, false, b,
                                                              (short)0, hc[t], false, false);
            }
        }
        // bias + tanh -> wave-private row-major [16][H] staging
        for (int t = 0; t < NT1; ++t) {
            float bias = b1s[16 * t + ncol];
            for (int r = 0; r < 8; ++r)
                hw[(mrow + r) * HDIM + 16 * t + ncol] = tanhf(hc[t][r] + bias);
        }
        __syncthreads();
        // GEMM2: res = h @ W2^T + b2   [16 x 64]
        v8f c4[NT2] = {v8f{}, v8f{}, v8f{}, v8f{}};
        for (int kc = 0; kc < KC2; ++kc) {
            v2f a = *(const v2f*)&hw[ncol * HDIM + 4 * kc + koff];
            for (int t = 0; t < NT2; ++t) {
                v2f b = *(const v2f*)&w2s[(16 * t + ncol) * HDIM + 4 * kc + koff];
                c4[t] = __builtin_amdgcn_wmma_f32_16x16x4_f32(false, a, false, b,
                                                              (short)0, c4[t], false, false);
            }
        }
        for (int t = 0; t < NT2; ++t) {
            float bias = b2s[16 * t + ncol];
            for (int r = 0; r < 8; ++r) res[t][r] = c4[t][r] + bias;
        }
    };

    // ---- classic RK4, one step per output interval (dt = 1/63) ----
    const float dt = 1.0f / (TPTS - 1);
    for (int ti = 1; ti < TPTS; ++ti) {
        stage_store(z, z, 0.0f);                 // stage = z
        vf(kd);                                  // k1
        for (int t = 0; t < NT2; ++t) for (int r = 0; r < 8; ++r) acc[t][r] = kd[t][r];

        stage_store(z, kd, 0.5f * dt);           // stage = z + dt/2*k1
        vf(kd);                                  // k2
        for (int t = 0; t < NT2; ++t) for (int r = 0; r < 8; ++r) acc[t][r] += 2.0f * kd[t][r];

        stage_store(z, kd, 0.5f * dt);           // stage = z + dt/2*k2
        vf(kd);                                  // k3
        for (int t = 0; t < NT2; ++t) for (int r = 0; r < 8; ++r) acc[t][r] += 2.0f * kd[t][r];

        stage_store(z, kd, dt);                  // stage = z + dt*k3
        vf(kd);                                  // k4
        for (int t = 0; t < NT2; ++t) for (int r = 0; r < 8; ++r) acc[t][r] += kd[t][r];

        for (int t = 0; t < NT2; ++t)
            for (int r = 0; r < 8; ++r)
                z[t][r] += (dt * (1.0f / 6.0f)) * acc[t][r];

        // write z[ti]
        float* zo = out + NDRAWS * DDIM + TPTS + (size_t)ti * NDRAWS * DDIM;
        for (int t = 0; t < NT2; ++t)
            for (int r = 0; r < 8; ++r)
                zo[(rowbase + mrow + r) * DDIM + 16 * t + ncol] = z[t][r];
    }
}

extern "C" void kernel_launch(void* const* d_in, const int* in_sizes, int n_in,
                              void* d_out, int out_size, void* d_ws, size_t ws_size,
                              hipStream_t stream) {
    const float* rand_e       = (const float*)d_in[0];
    const float* z0_mean      = (const float*)d_in[1];
    const float* z0_log_sigma = (const float*)d_in[2];
    const float* W1           = (const float*)d_in[3];
    const float* b1           = (const float*)d_in[4];
    const float* W2           = (const float*)d_in[5];
    const float* b2           = (const float*)d_in[6];
    float* out = (float*)d_out;
    (void)in_sizes; (void)n_in; (void)d_ws; (void)ws_size; (void)out_size;

    GenODE_node_rk4_wmma_f32<<<NBLOCKS, BLOCK, 0, stream>>>(
        rand_e, z0_mean, z0_log_sigma, W1, b1, W2, b2, out);
}